// CustomLSTM_82016695484538
// MI455X (gfx1250) — compile-verified
//
#include <hip/hip_runtime.h>

typedef float v2f __attribute__((ext_vector_type(2)));
typedef float v8f __attribute__((ext_vector_type(8)));

#define S_LEN 512
#define B_DIM 64
#define H_DIM 1024
#define FOUR_H 4096
#define K_DIM 1024

// Async global->LDS copy of 16B per lane; instruction offset is added to BOTH
// the global address (VADDR) and the LDS address (VDST) per the CDNA5 ISA.
#define ASYNC_CP16(ldsoff, gptr, OFFS)                                         \
    asm volatile("global_load_async_to_lds_b128 %0, %1, off offset:" OFFS     \
                 :: "v"(ldsoff), "v"(gptr) : "memory")
#define WAIT_ASYNC(N) asm volatile("s_wait_asynccnt " N ::: "memory")

__device__ __forceinline__ unsigned lds_off(const void* p) {
    // Low 32 bits of a generic pointer into LDS are the LDS byte address.
    return (unsigned)(size_t)p;
}

__device__ __forceinline__ float sigf(float x) { return 1.0f / (1.0f + __expf(-x)); }
__device__ __forceinline__ float tanh_fast(float x) { return 2.0f * sigf(2.0f * x) - 1.0f; }

// One LSTM time-step for one layer.
// gates = In @ Wih + h_prev @ Whh + bih + bhh  (M=64, N=4096, K=1024 each)
// Block: 16 hidden columns (j) x all 4 gate segments x all 64 batch rows.
// 4 waves; wave w owns M-tile rows [16w, 16w+16); acc = 4 segments x v8f.
__global__ __launch_bounds__(128) void lstm_step_kernel(
    const float* __restrict__ In,      // B x 1024 (x_t or h0_seq[t])
    const float* __restrict__ Wih,     // 1024 x 4H
    const float* __restrict__ bih,     // 4H
    const float* __restrict__ Whh,     // H x 4H
    const float* __restrict__ bhh,     // 4H
    const float* __restrict__ h_prev,  // B x H, nullptr when t==0
    float* __restrict__ c_state,       // B x H (in/out; ignored on read when t==0)
    float* __restrict__ h_out,         // B x H
    float* __restrict__ hn_dst,        // B x H, non-null only on last step
    float* __restrict__ cn_dst)        // B x H, non-null only on last step
{
    // Double-buffered tiles.
    // A: 64 rows x 32 k, stride 36 (16B-aligned rows, conflict-free ds_load_b64)
    __shared__ float Abuf[2][64 * 36];
    // W: 32 k x 64 cols (4 segs x 16), stride 80 (half-waves hit disjoint banks)
    __shared__ float Wbuf[2][32 * 80];

    const int tid    = threadIdx.x;
    const int wave   = tid >> 5;        // wave32
    const int lane   = tid & 31;
    const int half   = lane >> 4;       // 0: lanes 0-15, 1: lanes 16-31
    const int lane16 = lane & 15;
    const int jb     = blockIdx.x * 16; // hidden-column tile base
    const int m_tile = wave;
    const bool first = (h_prev == nullptr);

    v8f acc[4] = {};                    // segments i, f, o, g

    // Per-thread staging assignment (fixed across chunks):
    const int rowA = tid >> 1;          // A: 2 threads/row, 16 floats each
    const int ka   = (tid & 1) * 16;
    const int kw   = tid >> 2;          // W: thread -> (k row, one segment's 16 cols)
    const int seg  = tid & 3;

    const int nck = first ? 32 : 64;    // 32-wide K chunks over [ih | hh] phases

    // Issue async copies for chunk ck into LDS buffer bsel (8 x b128 per thread).
    auto issue = [&](int ck, int bsel) {
        const int phase = ck >> 5;
        const int k0    = (ck & 31) << 5;
        const float* Asrc = phase ? h_prev : In;
        const float* Wsrc = phase ? Whh : Wih;
        const float* ga = Asrc + (size_t)rowA * K_DIM + k0 + ka;
        const unsigned la = lds_off(&Abuf[bsel][rowA * 36 + ka]);
        ASYNC_CP16(la, ga, "0");
        ASYNC_CP16(la, ga, "16");
        ASYNC_CP16(la, ga, "32");
        ASYNC_CP16(la, ga, "48");
        const float* gw = Wsrc + (size_t)(k0 + kw) * FOUR_H + seg * H_DIM + jb;
        const unsigned lw = lds_off(&Wbuf[bsel][kw * 80 + seg * 16]);
        ASYNC_CP16(lw, gw, "0");
        ASYNC_CP16(lw, gw, "16");
        ASYNC_CP16(lw, gw, "32");
        ASYNC_CP16(lw, gw, "48");
    };

    issue(0, 0);
    issue(1, 1);

    for (int ck = 0; ck < nck; ++ck) {
        // In-order async completion: leaving <=8 outstanding means chunk ck landed.
        if (ck + 1 < nck) { WAIT_ASYNC("0x8"); } else { WAIT_ASYNC("0x0"); }
        __syncthreads();

        const int bsel = ck & 1;
        const float* Ab = &Abuf[bsel][0];
        const float* Wb = &Wbuf[bsel][0];
        const int arow = (16 * m_tile + lane16) * 36;
        #pragma unroll
        for (int kk = 0; kk < 32; kk += 4) {
            // A frag (16x4 f32): lanes 0-15 -> K=kk,kk+1 ; lanes 16-31 -> K=kk+2,kk+3
            const int kA = kk + 2 * half;
            v2f a = *(const v2f*)(&Ab[arow + kA]);   // ds_load_b64
            #pragma unroll
            for (int s = 0; s < 4; ++s) {
                // B frag (4x16 f32): VGPR0 = K(kA), VGPR1 = K(kA+1), N = lane16
                v2f b;
                b.x = Wb[kA * 80 + s * 16 + lane16];
                b.y = Wb[(kA + 1) * 80 + s * 16 + lane16];
                acc[s] = __builtin_amdgcn_wmma_f32_16x16x4_f32(
                    false, a, false, b, (short)0, acc[s], false, false);
            }
        }

        __syncthreads();                 // WAR: all waves done reading buf[ck&1]
        if (ck + 2 < nck) issue(ck + 2, bsel);
    }

    // ---- LSTM cell (elementwise), using C-layout: row M = r + 8*half, col N = lane16
    const int j  = jb + lane16;
    const float bI = bih[j]            + bhh[j];
    const float bF = bih[H_DIM + j]    + bhh[H_DIM + j];
    const float bO = bih[2*H_DIM + j]  + bhh[2*H_DIM + j];
    const float bG = bih[3*H_DIM + j]  + bhh[3*H_DIM + j];
    #pragma unroll
    for (int r = 0; r < 8; ++r) {
        const int m   = 16 * m_tile + r + 8 * half;
        const int idx = m * H_DIM + j;
        const float gi = acc[0][r] + bI;
        const float gf = acc[1][r] + bF;
        const float go = acc[2][r] + bO;
        const float gg = acc[3][r] + bG;
        const float cp = first ? 0.0f : c_state[idx];
        const float c  = sigf(gf) * cp + sigf(gi) * tanh_fast(gg);
        const float h  = sigf(go) * tanh_fast(c);
        c_state[idx] = c;
        h_out[idx]   = h;
        if (hn_dst) { hn_dst[idx] = h; cn_dst[idx] = c; }
    }
}

extern "C" void kernel_launch(void* const* d_in, const int* in_sizes, int n_in,
                              void* d_out, int out_size, void* d_ws, size_t ws_size,
                              hipStream_t stream) {
    (void)in_sizes; (void)n_in; (void)out_size; (void)ws_size;
    const float* x    = (const float*)d_in[0];
    const float* Wih0 = (const float*)d_in[1];
    const float* bih0 = (const float*)d_in[2];
    const float* Whh0 = (const float*)d_in[3];
    const float* bhh0 = (const float*)d_in[4];
    const float* Wih1 = (const float*)d_in[5];
    const float* bih1 = (const float*)d_in[6];
    const float* Whh1 = (const float*)d_in[7];
    const float* bhh1 = (const float*)d_in[8];
    float* out = (float*)d_out;

    const size_t bh = (size_t)B_DIM * H_DIM;
    // Workspace: layer-0 h sequence (S*B*H) + two cell states (B*H each) ~ 135 MB
    float* h0seq = (float*)d_ws;
    float* c0    = h0seq + (size_t)S_LEN * bh;
    float* c1    = c0 + bh;

    // d_out layout: output (S,B,H) | h_n (2,B,H) | c_n (2,B,H)
    float* out_seq = out;
    float* hn0 = out + (size_t)S_LEN * bh;
    float* hn1 = hn0 + bh;
    float* cn0 = hn1 + bh;
    float* cn1 = cn0 + bh;

    dim3 grid(H_DIM / 16), block(128);

    // Layer 0: reads x, writes h sequence to workspace
    for (int t = 0; t < S_LEN; ++t) {
        const bool last = (t == S_LEN - 1);
        lstm_step_kernel<<<grid, block, 0, stream>>>(
            x + (size_t)t * B_DIM * K_DIM, Wih0, bih0, Whh0, bhh0,
            t ? (h0seq + (size_t)(t - 1) * bh) : nullptr,
            c0, h0seq + (size_t)t * bh,
            last ? hn0 : nullptr, last ? cn0 : nullptr);
    }
    // Layer 1: reads layer-0 h sequence, writes final output sequence
    for (int t = 0; t < S_LEN; ++t) {
        const bool last = (t == S_LEN - 1);
        lstm_step_kernel<<<grid, block, 0, stream>>>(
            h0seq + (size_t)t * bh, Wih1, bih1, Whh1, bhh1,
            t ? (out_seq + (size_t)(t - 1) * bh) : nullptr,
            c1, out_seq + (size_t)t * bh,
            last ? hn1 : nullptr, last ? cn1 : nullptr);
    }
}